// JGCF_52707838657040
// MI455X (gfx1250) — compile-verified
//
#include <hip/hip_runtime.h>
#include <math.h>

#define N_USERS 100000
#define M_ITEMS 50000
#define N_ALL   150000
#define N_EDGES 4800000
#define DIM     64
#define NCH     ((N_ALL + 255) / 256)   // 586 chunks for the scan

typedef __attribute__((ext_vector_type(4))) float v4f;   // clang vector for nontemporal builtins

// ---------------- CDNA5 async-to-LDS helpers (gfx1250) ----------------
// Non-temporal: edge stream is read once per layer; keep L2 for the gather arrays.
__device__ __forceinline__ void async_load_b32_nt(unsigned lds_addr, const void* sbase, int byte_off) {
  // GVS mode: mem = SADDR(64) + VADDR[31:0] ; VDST = LDS byte address
  asm volatile("global_load_async_to_lds_b32 %0, %1, %2 th:TH_LOAD_NT"
               :: "v"(lds_addr), "v"(byte_off), "s"(sbase)
               : "memory");
}
__device__ __forceinline__ void wait_asynccnt0() {
  asm volatile("s_wait_asynccnt 0" ::: "memory");
}

// ---------------- CSR build ----------------
__global__ void k_hist(const int* __restrict__ rows, int* __restrict__ counts) {
  int e = blockIdx.x * 256 + threadIdx.x;
  if (e < N_EDGES) atomicAdd(&counts[rows[e]], 1);
}

__global__ void k_chunksum(const int* __restrict__ counts, int* __restrict__ csum) {
  __shared__ int sm[256];
  int i = blockIdx.x * 256 + threadIdx.x;
  int v = (i < N_ALL) ? counts[i] : 0;
  sm[threadIdx.x] = v;
  __syncthreads();
  for (int o = 128; o > 0; o >>= 1) {
    if (threadIdx.x < o) sm[threadIdx.x] += sm[threadIdx.x + o];
    __syncthreads();
  }
  if (threadIdx.x == 0) csum[blockIdx.x] = sm[0];
}

__global__ void k_scanchunks(const int* __restrict__ csum, int* __restrict__ coff) {
  __shared__ int sm[1024];
  int t = threadIdx.x;
  int v = (t < NCH) ? csum[t] : 0;
  sm[t] = v;
  __syncthreads();
  for (int o = 1; o < 1024; o <<= 1) {
    int x = (t >= o) ? sm[t - o] : 0;
    __syncthreads();
    sm[t] += x;
    __syncthreads();
  }
  if (t < NCH) coff[t] = sm[t] - v;   // exclusive
}

__global__ void k_scanlocal(const int* __restrict__ counts, const int* __restrict__ coff,
                            int* __restrict__ rowStart, int* __restrict__ cursor) {
  __shared__ int sm[256];
  int t = threadIdx.x;
  int i = blockIdx.x * 256 + t;
  int v = (i < N_ALL) ? counts[i] : 0;
  sm[t] = v;
  __syncthreads();
  for (int o = 1; o < 256; o <<= 1) {
    int x = (t >= o) ? sm[t - o] : 0;
    __syncthreads();
    sm[t] += x;
    __syncthreads();
  }
  if (i < N_ALL) {
    int ex = coff[blockIdx.x] + sm[t] - v;   // exclusive prefix
    rowStart[i] = ex;
    cursor[i]   = ex;
  }
}

__global__ void k_scatter(const int* __restrict__ rows, const int* __restrict__ cols,
                          const float* __restrict__ vals, int* __restrict__ cursor,
                          int* __restrict__ ccol, float* __restrict__ cval) {
  int e = blockIdx.x * 256 + threadIdx.x;
  if (e >= N_EDGES) return;
  int r = rows[e];
  int p = atomicAdd(&cursor[r], 1);
  ccol[p] = cols[e];
  cval[p] = vals[e];
}

// ---------------- embed build: B0 = concat(user,item); S = B0 ----------------
__global__ void k_build(const float* __restrict__ ue, const float* __restrict__ ie,
                        float* __restrict__ B0, float* __restrict__ S) {
  int i = blockIdx.x * 256 + threadIdx.x;        // float4 index
  if (i >= N_ALL * 16) return;
  int r = i >> 4, d4 = i & 15;
  float4 v = (r < N_USERS) ? ((const float4*)ue)[i]
                           : ((const float4*)ie)[(size_t)(r - N_USERS) * 16 + d4];
  ((float4*)B0)[i] = v;
  ((float4*)S)[i]  = v;
}

// ---------------- fused Jacobi layer: dst = c_acc*spmm(src) + c_lin*lin; S += dst ----------------
// wave-per-row; two 16-lane edge pipelines, 4x unrolled (8 gathers in flight per wave);
// (col,val) staged to LDS via CDNA5 async loads (ASYNCcnt), read back via DS broadcast.
__global__ void __launch_bounds__(256)
k_jacobi(const float* __restrict__ src, const float* __restrict__ lin,
         float* __restrict__ dst,              // may be null (layer 3)
         float* __restrict__ S,
         const int* __restrict__ rowStart, const int* __restrict__ counts,
         const int* __restrict__ ccol, const float* __restrict__ cval,
         float c_acc, float c_lin) {
  __shared__ int lds[8 * 64];                   // per-wave: 32 cols + 32 vals
  const int wib  = threadIdx.x >> 5;
  const int lane = threadIdx.x & 31;
  const int row  = blockIdx.x * 8 + wib;
  if (row >= N_ALL) return;                     // wave-uniform
  const int half = lane >> 4;
  const int sub  = lane & 15;

  const int start = rowStart[row];
  const int len   = counts[row];

  int* mc = &lds[wib * 64];
  int* mv = mc + 32;
  const unsigned laddrC = (unsigned)(size_t)(mc + lane);
  const unsigned laddrV = (unsigned)(size_t)(mv + lane);

  float4 acc0 = make_float4(0.f, 0.f, 0.f, 0.f);
  float4 acc1 = make_float4(0.f, 0.f, 0.f, 0.f);

  for (int ch = 0; ch < len; ch += 32) {        // wave-uniform loop
    int n = len - ch; if (n > 32) n = 32;
    if (lane < n) {
      int boff = (start + ch + lane) * 4;
      async_load_b32_nt(laddrC, (const void*)ccol, boff);
      async_load_b32_nt(laddrV, (const void*)cval, boff);
    }
    wait_asynccnt0();

    int i = half;
    // 4 edges per half in flight: cols/vals from LDS, 4 independent 256B gathers
    for (; i + 6 < n; i += 8) {
      int   c0 = mc[i],     c1 = mc[i + 2],  c2 = mc[i + 4],  c3 = mc[i + 6];
      float v0 = __int_as_float(mv[i]);
      float v1 = __int_as_float(mv[i + 2]);
      float v2 = __int_as_float(mv[i + 4]);
      float v3 = __int_as_float(mv[i + 6]);
      float4 x0 = *((const float4*)(src + (size_t)c0 * DIM) + sub);
      float4 x1 = *((const float4*)(src + (size_t)c1 * DIM) + sub);
      float4 x2 = *((const float4*)(src + (size_t)c2 * DIM) + sub);
      float4 x3 = *((const float4*)(src + (size_t)c3 * DIM) + sub);
      acc0.x = fmaf(v0, x0.x, acc0.x); acc0.y = fmaf(v0, x0.y, acc0.y);
      acc0.z = fmaf(v0, x0.z, acc0.z); acc0.w = fmaf(v0, x0.w, acc0.w);
      acc1.x = fmaf(v1, x1.x, acc1.x); acc1.y = fmaf(v1, x1.y, acc1.y);
      acc1.z = fmaf(v1, x1.z, acc1.z); acc1.w = fmaf(v1, x1.w, acc1.w);
      acc0.x = fmaf(v2, x2.x, acc0.x); acc0.y = fmaf(v2, x2.y, acc0.y);
      acc0.z = fmaf(v2, x2.z, acc0.z); acc0.w = fmaf(v2, x2.w, acc0.w);
      acc1.x = fmaf(v3, x3.x, acc1.x); acc1.y = fmaf(v3, x3.y, acc1.y);
      acc1.z = fmaf(v3, x3.z, acc1.z); acc1.w = fmaf(v3, x3.w, acc1.w);
    }
    for (; i < n; i += 2) {
      int   c = mc[i];
      float v = __int_as_float(mv[i]);
      float4 x = *((const float4*)(src + (size_t)c * DIM) + sub);
      acc0.x = fmaf(v, x.x, acc0.x); acc0.y = fmaf(v, x.y, acc0.y);
      acc0.z = fmaf(v, x.z, acc0.z); acc0.w = fmaf(v, x.w, acc0.w);
    }
  }

  float4 acc;
  acc.x = acc0.x + acc1.x; acc.y = acc0.y + acc1.y;
  acc.z = acc0.z + acc1.z; acc.w = acc0.w + acc1.w;

  // combine the two 16-lane halves (same dims, different edges)
  acc.x += __shfl_xor(acc.x, 16, 32);
  acc.y += __shfl_xor(acc.y, 16, 32);
  acc.z += __shfl_xor(acc.z, 16, 32);
  acc.w += __shfl_xor(acc.w, 16, 32);

  if (half == 0) {
    size_t o = (size_t)row * DIM + (size_t)sub * 4;
    float4 l = *(const float4*)(lin + o);
    float4 d;
    d.x = fmaf(c_acc, acc.x, c_lin * l.x);
    d.y = fmaf(c_acc, acc.y, c_lin * l.y);
    d.z = fmaf(c_acc, acc.z, c_lin * l.z);
    d.w = fmaf(c_acc, acc.w, c_lin * l.w);
    if (dst) *(float4*)(dst + o) = d;
    float4 s = *(float4*)(S + o);
    s.x += d.x; s.y += d.y; s.z += d.z; s.w += d.w;
    *(float4*)(S + o) = s;
  }
}

// ---------------- finalize: band_stop = S/4 ; band_pass = tanh(0.1*e0 - band_stop) ----------------
__global__ void k_finalize(const float* __restrict__ B0, const float* __restrict__ S,
                           float* __restrict__ out) {
  int i = blockIdx.x * 256 + threadIdx.x;       // float4 index over N_ALL*16
  if (i >= N_ALL * 16) return;
  int r = i >> 4, d4 = i & 15;
  float4 s = ((const float4*)S)[i];
  float4 e = ((const float4*)B0)[i];
  v4f bs, bp;
  bs.x = s.x * 0.25f; bs.y = s.y * 0.25f; bs.z = s.z * 0.25f; bs.w = s.w * 0.25f;
  bp.x = tanhf(fmaf(0.1f, e.x, -bs.x));
  bp.y = tanhf(fmaf(0.1f, e.y, -bs.y));
  bp.z = tanhf(fmaf(0.1f, e.z, -bs.z));
  bp.w = tanhf(fmaf(0.1f, e.w, -bs.w));
  // output written once, never read again -> non-temporal stores keep L2 clean
  float* orow = out + (size_t)r * 128;
  __builtin_nontemporal_store(bs, (v4f*)(orow + (size_t)d4 * 4));          // dims 0..63
  __builtin_nontemporal_store(bp, (v4f*)(orow + 64 + (size_t)d4 * 4));     // dims 64..127
}

// ---------------- host ----------------
extern "C" void kernel_launch(void* const* d_in, const int* in_sizes, int n_in,
                              void* d_out, int out_size, void* d_ws, size_t ws_size,
                              hipStream_t stream) {
  const float* ue   = (const float*)d_in[0];
  const float* ie   = (const float*)d_in[1];
  const int*   rows = (const int*)d_in[2];
  const int*   cols = (const int*)d_in[3];
  const float* vals = (const float*)d_in[4];
  float* out = (float*)d_out;

  char* w = (char*)d_ws;
  auto alloc = [&](size_t bytes) -> void* {
    void* p = (void*)w;
    w += (bytes + 255) & ~(size_t)255;
    return p;
  };
  const size_t EMB = (size_t)N_ALL * DIM * sizeof(float);   // 38.4 MB
  float* B0 = (float*)alloc(EMB);
  float* S  = (float*)alloc(EMB);
  float* B1 = (float*)alloc(EMB);
  float* B2 = (float*)alloc(EMB);
  int* counts   = (int*)alloc((size_t)N_ALL * 4);
  int* rowStart = (int*)alloc((size_t)N_ALL * 4);
  int* cursor   = (int*)alloc((size_t)N_ALL * 4);
  int* csum     = (int*)alloc((size_t)NCH * 4);
  int* coff     = (int*)alloc((size_t)NCH * 4);
  int*   ccol   = (int*)alloc((size_t)N_EDGES * 4);
  float* cval   = (float*)alloc((size_t)N_EDGES * 4);

  const int EB = (N_EDGES + 255) / 256;        // edge-grid blocks
  const int VB = (N_ALL * 16 + 255) / 256;     // float4-grid blocks
  const int RB = (N_ALL + 7) / 8;              // wave-per-row blocks (8 waves/block)

  // CSR build (stateless, every call)
  (void)hipMemsetAsync(counts, 0, (size_t)N_ALL * 4, stream);
  k_hist<<<EB, 256, 0, stream>>>(rows, counts);
  k_chunksum<<<NCH, 256, 0, stream>>>(counts, csum);
  k_scanchunks<<<1, 1024, 0, stream>>>(csum, coff);
  k_scanlocal<<<NCH, 256, 0, stream>>>(counts, coff, rowStart, cursor);
  k_scatter<<<EB, 256, 0, stream>>>(rows, cols, vals, cursor, ccol, cval);

  // e0 and running sum
  k_build<<<VB, 256, 0, stream>>>(ue, ie, B0, S);

  // Jacobi layers (A=B=1): theta2 == 0 everywhere
  // k=1: e1 = 1.0*spmm(e0) + 0.0*e0
  k_jacobi<<<RB, 256, 0, stream>>>(B0, B0, B1, S, rowStart, counts, ccol, cval,
                                   1.0f, 0.0f);
  // k=2: e2 = 1.875*spmm(e1) - 0.75*e0
  k_jacobi<<<RB, 256, 0, stream>>>(B1, B0, B2, S, rowStart, counts, ccol, cval,
                                   1.875f, -0.75f);
  // k=3: e3 = (56/30)*spmm(e2) - 0.8*e1   (accumulated straight into S)
  k_jacobi<<<RB, 256, 0, stream>>>(B2, B1, nullptr, S, rowStart, counts, ccol, cval,
                                   (float)(56.0 / 30.0), -0.8f);

  k_finalize<<<VB, 256, 0, stream>>>(B0, S, out);
}